// SWD9_28449863369553
// MI455X (gfx1250) — compile-verified
//
#include <hip/hip_runtime.h>
#include <stdint.h>

// Problem shape (fixed by the reference): B=4, H=8, N=2048, DV=64.
#define B_   4
#define H_   8
#define N_   2048
#define DV_  64
#define BH_  (B_ * H_)          // 32
#define NCOL (BH_ * DV_)        // 2048 columns of length N to sort

#define AS1 __attribute__((address_space(1)))
#define AS3 __attribute__((address_space(3)))

typedef uint32_t u32x4 __attribute__((ext_vector_type(4)));

// ---------------------------------------------------------------------------
// CDNA5 async global->LDS copy (ASYNCcnt path, CDNA5 ISA)
// ---------------------------------------------------------------------------
static __device__ __forceinline__ void async_copy_b32_to_lds(const void* g, void* lds) {
#if __has_builtin(__builtin_amdgcn_global_load_async_to_lds_b32)
  __builtin_amdgcn_global_load_async_to_lds_b32(
      (AS1 int*)(g), (AS3 int*)(lds), /*offset=*/0, /*cpol=*/0);
#else
  // GV mode: VDST = LDS byte offset (low 32 bits of the shared-aperture
  // address), VADDR = 64-bit global address.
  uint32_t l = (uint32_t)(uintptr_t)lds;
  uint64_t a = (uint64_t)(uintptr_t)g;
  asm volatile("global_load_async_to_lds_b32 %0, %1, off"
               :: "v"(l), "v"(a) : "memory");
#endif
}

static __device__ __forceinline__ void wait_asynccnt0() {
#if __has_builtin(__builtin_amdgcn_s_wait_asynccnt)
  __builtin_amdgcn_s_wait_asynccnt(0);
#else
  asm volatile("s_wait_asynccnt 0" ::: "memory");
#endif
}

// ---------------------------------------------------------------------------
// Kernel 1: per-column bitonic sort in LDS.
// One block per column c = bh*DV + d. Sorts by composite key (value, index)
// ascending -> identical result to stable argsort. Writes P_d (perm) and
// P_d^{-1} (inv) to workspace.
// ---------------------------------------------------------------------------
__global__ __launch_bounds__(256) void sort_cols_kernel(
    const float* __restrict__ v, int* __restrict__ perm, int* __restrict__ inv) {
  __shared__ float key[N_];
  __shared__ int   idx[N_];

  const int c  = blockIdx.x;
  const int bh = c / DV_;
  const int d  = c - bh * DV_;
  const int t  = threadIdx.x;
  const float* col = v + (size_t)bh * N_ * DV_ + d;   // stride DV_ elements

  // Async strided column load straight into LDS (no VGPR round trip).
  for (int e = t; e < N_; e += 256) {
    async_copy_b32_to_lds(col + (size_t)e * DV_, &key[e]);
    idx[e] = e;
  }
  wait_asynccnt0();
  __syncthreads();

  // Bitonic sort: 66 passes, 1024 disjoint compare-exchanges per pass,
  // 4 per thread per pass.
  for (int k = 2; k <= N_; k <<= 1) {
    for (int j = k >> 1; j > 0; j >>= 1) {
      for (int p = t; p < N_ / 2; p += 256) {
        const int  i  = p + (p & ~(j - 1));   // insert 0 bit at position log2(j)
        const int  l  = i | j;
        const bool up = ((i & k) == 0);
        const float ki = key[i], kl = key[l];
        const int   ii = idx[i], il = idx[l];
        const bool  lt = (ki < kl) || (ki == kl && ii < il);  // strict total order
        if (lt != up) { key[i] = kl; key[l] = ki; idx[i] = il; idx[l] = ii; }
      }
      __syncthreads();
    }
  }

  int* permc = perm + (size_t)c * N_;
  int* invc  = inv  + (size_t)c * N_;
  for (int e = t; e < N_; e += 256) {
    const int r = idx[e];
    permc[e] = r;   // P_d[e]   : row of e-th smallest
    invc[r]  = e;   // P_d^{-1}[r] : rank of row r
  }
}

// ---------------------------------------------------------------------------
// Kernel 2: out[bh,i,d] = v[bh, P_d[ P^{-1}_{(d+1)%DV}[i] ], d]
// All tables are L2-resident (v 16MB + perm/inv 32MB << 192MB L2).
// ---------------------------------------------------------------------------
__global__ __launch_bounds__(256) void out_kernel(
    const float* __restrict__ v, const int* __restrict__ perm,
    const int* __restrict__ inv, float* __restrict__ out) {
  const size_t g  = (size_t)blockIdx.x * 256 + threadIdx.x;   // < BH*N*DV
  const int d  = (int)(g & (DV_ - 1));
  const int i  = (int)((g >> 6) & (N_ - 1));
  const int bh = (int)(g >> 17);                               // 6 + 11 bits
  const int dn = (d + 1) & (DV_ - 1);
  const int j  = inv [(size_t)(bh * DV_ + dn) * N_ + i];
  const int r  = perm[(size_t)(bh * DV_ + d ) * N_ + j];
  out[g] = v[((size_t)bh * N_ + r) * DV_ + d];
}

// ---------------------------------------------------------------------------
// Kernel 3: stream-zero the 512MB attn buffer with non-temporal 128-bit
// stores (bigger than L2 -> NT avoids evicting the hot index tables).
// ---------------------------------------------------------------------------
__global__ __launch_bounds__(256) void zero_attn_kernel(u32x4* __restrict__ p, size_t n16) {
  const size_t stride = (size_t)gridDim.x * 256;
  const u32x4 z = {0u, 0u, 0u, 0u};
  for (size_t i = (size_t)blockIdx.x * 256 + threadIdx.x; i < n16; i += stride)
    __builtin_nontemporal_store(z, p + i);
}

// ---------------------------------------------------------------------------
// Kernel 4: attn[bh, i, P_0[P^{-1}_1[i]]] = 1  (int32 one-hot)
// ---------------------------------------------------------------------------
__global__ __launch_bounds__(256) void scatter_ones_kernel(
    const int* __restrict__ perm, const int* __restrict__ inv, int* __restrict__ attn) {
  const int g  = blockIdx.x * 256 + threadIdx.x;               // < BH*N
  const int i  = g & (N_ - 1);
  const int bh = g >> 11;
  const int j  = inv [(size_t)(bh * DV_ + 1) * N_ + i];
  const int a  = perm[(size_t)(bh * DV_    ) * N_ + j];
  attn[((size_t)bh * N_ + i) * N_ + a] = 1;
}

// ---------------------------------------------------------------------------
extern "C" void kernel_launch(void* const* d_in, const int* in_sizes, int n_in,
                              void* d_out, int out_size, void* d_ws, size_t ws_size,
                              hipStream_t stream) {
  (void)in_sizes; (void)n_in; (void)out_size; (void)ws_size;

  // setup_inputs order: q, k, v -- only v is used by the reference.
  const float* v = (const float*)d_in[2];

  // d_out = [ out : BH*N*DV f32 | attn : BH*N*N i32 ] (tuple concat, both 4B).
  float* out  = (float*)d_out;
  int*   attn = (int*)(out + (size_t)BH_ * N_ * DV_);

  // Workspace: perm (16MB) + inv (16MB).
  int* perm = (int*)d_ws;
  int* inv  = perm + (size_t)NCOL * N_;

  sort_cols_kernel<<<NCOL, 256, 0, stream>>>(v, perm, inv);
  out_kernel<<<(BH_ * N_ * DV_) / 256, 256, 0, stream>>>(v, perm, inv, out);

  const size_t n16 = ((size_t)BH_ * N_ * N_) / 4;              // 128-bit chunks
  zero_attn_kernel<<<16384, 256, 0, stream>>>((u32x4*)attn, n16);
  scatter_ones_kernel<<<(BH_ * N_) / 256, 256, 0, stream>>>(perm, inv, attn);
}